// HierarchicalSparseModel_11527692222862
// MI455X (gfx1250) — compile-verified
//
#include <hip/hip_runtime.h>
#include <stdint.h>

// ---------------------------------------------------------------------------
// MI455X (gfx1250) implementation. Memory-bound workload (~1.3GB traffic,
// ~1 TFLOP fp32) -> keep fp32 end-to-end and use V_WMMA_F32_16X16X4_F32 for
// all large GEMMs. Shared tiles (weights / activations reused by multiple
// waves of a workgroup) are staged into LDS with GLOBAL_LOAD_ASYNC_TO_LDS_B128
// (ASYNCcnt + double buffering) to remove the 2-4x redundant L2 traffic that
// would otherwise sit on top of the 23.3 TB/s HBM floor.
// ---------------------------------------------------------------------------

typedef float v2f __attribute__((ext_vector_type(2)));
typedef float v8f __attribute__((ext_vector_type(8)));

static constexpr int B_ = 4, T_ = 1024, D_ = 1024, L_ = 16;
static constexpr int CD_ = 256, C_ = 64, RD_ = 128, FD_ = 4096, V_ = 32000;
static constexpr int NR_ = 7, DP_ = 9, M_ = B_ * T_;   // 4096 rows
static constexpr int NBLK = 64;                        // T/16 token blocks

#define DEV __device__ __forceinline__

DEV float geluf(float x) { return 0.5f * x * (1.0f + erff(x * 0.70710678118654752440f)); }
DEV float sigmf(float x) { return 1.0f / (1.0f + expf(-x)); }

DEV v8f wmma_f32(v2f a, v2f b, v8f c) {
  // D(16x16,f32) = A(16x4,f32) x B(4x16,f32) + C
  return __builtin_amdgcn_wmma_f32_16x16x4_f32(false, a, false, b, (short)0, c,
                                               false, false);
}

// Async copy of 16 bytes global -> LDS (tracked by ASYNCcnt).
DEV void async_cp16(float* lds_dst, const float* gsrc) {
  unsigned ldsa = (unsigned)(size_t)lds_dst;  // flat LDS aperture: addr[31:0]
  asm volatile("global_load_async_to_lds_b128 %0, %1, off"
               :: "v"(ldsa), "v"(gsrc) : "memory");
}
DEV void async_wait0() {
  asm volatile("s_wait_asynccnt 0x0" ::: "memory");
}

// ---------------------------------------------------------------------------
// Small dense helpers (controller / router MLPs: negligible FLOPs)
// ---------------------------------------------------------------------------

__global__ void k_summaries(const float* __restrict__ dense, float* __restrict__ sums) {
  int idx = blockIdx.x * blockDim.x + threadIdx.x;
  if (idx >= 3 * B_ * D_) return;
  int d = idx % D_;
  int b = (idx / D_) % B_;
  int s = idx / (B_ * D_);
  int layer = (s == 0) ? 1 : (s == 1 ? 5 : 9);
  const float* p = dense + ((size_t)layer * B_ + b) * (size_t)T_ * D_ + d;
  float acc = 0.f;
  for (int t = 0; t < T_; ++t) acc += p[(size_t)t * D_];
  sums[idx] = acc * (1.0f / T_);
}

// y[row][o] = act(sum_k x[row*ldx + k] * w[o*K + k] + bias[o])
__global__ void k_linear(const float* __restrict__ x, int ldx,
                         const float* __restrict__ w, const float* __restrict__ bias,
                         float* __restrict__ y, int rows, int K, int OUT, int act) {
  int idx = blockIdx.x * blockDim.x + threadIdx.x;
  if (idx >= rows * OUT) return;
  int row = idx / OUT, o = idx % OUT;
  const float* xr = x + (size_t)row * ldx;
  const float* wr = w + (size_t)o * K;
  float acc = bias ? bias[o] : 0.f;
  for (int k = 0; k < K; ++k) acc = fmaf(xr[k], wr[k], acc);
  if (act == 1) acc = geluf(acc);
  else if (act == 2) acc = sigmf(acc);
  y[(size_t)row * OUT + o] = acc;
}

__global__ void k_proj_concat(const float* __restrict__ sums, const float* __restrict__ w,
                              const float* __restrict__ bias, float* __restrict__ pcat) {
  int idx = blockIdx.x * blockDim.x + threadIdx.x;
  if (idx >= B_ * 3 * CD_) return;
  int o = idx % (3 * CD_);
  int b = idx / (3 * CD_);
  int s = o / CD_, oo = o % CD_;
  const float* xr = sums + ((size_t)s * B_ + b) * D_;
  const float* wr = w + (size_t)oo * D_;
  float acc = bias[oo];
  for (int k = 0; k < D_; ++k) acc = fmaf(xr[k], wr[k], acc);
  pcat[(size_t)b * (3 * CD_) + o] = acc;
}

__global__ void k_hcat(const float* __restrict__ g, const float* __restrict__ lq,
                       float* __restrict__ hcat) {
  int idx = blockIdx.x * blockDim.x + threadIdx.x;
  if (idx >= B_ * L_ * 2 * CD_) return;
  int col = idx % (2 * CD_);
  int row = idx / (2 * CD_);
  int b = row / L_, l = row % L_;
  hcat[idx] = (col < CD_) ? g[b * CD_ + col] : lq[l * CD_ + (col - CD_)];
}

__global__ void k_pool(const float* __restrict__ h, float* __restrict__ hp) {
  int idx = blockIdx.x * blockDim.x + threadIdx.x;
  if (idx >= B_ * NBLK * D_) return;
  int d = idx % D_;
  int tb = (idx / D_) % NBLK;
  int b = idx / (D_ * NBLK);
  const float* p = h + ((size_t)b * T_ + tb * 16) * (size_t)D_ + d;
  float a = 0.f;
  for (int r = 0; r < 16; ++r) a += p[(size_t)r * D_];
  hp[idx] = a * (1.0f / 16.0f);
}

// LayerNorm(row of 1024) then local projection (128 outs); one block per row.
__global__ void k_ln_local(const float* __restrict__ hp, const float* __restrict__ lng,
                           const float* __restrict__ lnb, const float* __restrict__ lw,
                           const float* __restrict__ lb, float* __restrict__ xl) {
  __shared__ float sh[D_];
  __shared__ float red[128];
  int row = blockIdx.x, tid = threadIdx.x;
  const float* xr = hp + (size_t)row * D_;
  float s = 0.f, s2 = 0.f;
  for (int k = tid; k < D_; k += 128) {
    float v = xr[k];
    sh[k] = v; s += v; s2 += v * v;
  }
  red[tid] = s; __syncthreads();
  for (int st = 64; st > 0; st >>= 1) { if (tid < st) red[tid] += red[tid + st]; __syncthreads(); }
  float mu = red[0] * (1.0f / D_); __syncthreads();
  red[tid] = s2; __syncthreads();
  for (int st = 64; st > 0; st >>= 1) { if (tid < st) red[tid] += red[tid + st]; __syncthreads(); }
  float var = red[0] * (1.0f / D_) - mu * mu; __syncthreads();
  float rstd = rsqrtf(var + 1e-5f);
  const float* wr = lw + (size_t)tid * D_;
  float acc = lb[tid];
  for (int k = 0; k < D_; ++k) {
    float nv = (sh[k] - mu) * rstd * lng[k] + lnb[k];
    acc = fmaf(nv, wr[k], acc);
  }
  xl[(size_t)row * RD_ + tid] = acc;
}

__global__ void k_router_vec(const float* __restrict__ cv, const float* __restrict__ fb,
                             const float* __restrict__ ss,
                             const float* __restrict__ cw, const float* __restrict__ cb,
                             const float* __restrict__ bw, const float* __restrict__ bb,
                             int li, float* __restrict__ xc, float* __restrict__ xb) {
  int idx = blockIdx.x * blockDim.x + threadIdx.x;
  if (idx >= B_ * RD_) return;
  int o = idx % RD_, b = idx / RD_;
  const float* cvr = cv + ((size_t)b * L_ + li) * C_;
  const float* wr = cw + (size_t)o * C_;
  float a = cb[o];
  for (int c = 0; c < C_; ++c) a = fmaf(cvr[c], wr[c], a);
  xc[idx] = a;
  float fbv = fb[b * L_ + li], ssv = ss[b * L_ + li];
  xb[idx] = bw[o * 2 + 0] * fbv + bw[o * 2 + 1] * ssv + bb[o];
}

__global__ void k_trunk1(const float* __restrict__ xl, const float* __restrict__ xc,
                         const float* __restrict__ xb, const float* __restrict__ w,
                         const float* __restrict__ bias, float* __restrict__ t1) {
  int idx = blockIdx.x * blockDim.x + threadIdx.x;
  if (idx >= B_ * NBLK * RD_) return;
  int o = idx % RD_;
  int row = idx / RD_;
  int b = row / NBLK;
  const float* wr = w + (size_t)o * (3 * RD_);
  const float* x0 = xl + (size_t)row * RD_;
  const float* x1 = xc + (size_t)b * RD_;
  const float* x2 = xb + (size_t)b * RD_;
  float a = bias[o];
  for (int k = 0; k < RD_; ++k) a = fmaf(x0[k], wr[k], a);
  for (int k = 0; k < RD_; ++k) a = fmaf(x1[k], wr[RD_ + k], a);
  for (int k = 0; k < RD_; ++k) a = fmaf(x2[k], wr[2 * RD_ + k], a);
  t1[idx] = geluf(a);
}

// k = int(mean_b(clip(rint(fb*4096), 1, 4096)))  (rintf == RNE == jnp.round)
__global__ void k_kcalc(const float* __restrict__ fb, int li, int* __restrict__ kout) {
  float s = 0.f;
  for (int b = 0; b < B_; ++b) {
    float v = rintf(fb[b * L_ + li] * 4096.0f);
    v = fminf(fmaxf(v, 1.0f), 4096.0f);
    s += v;
  }
  kout[0] = (int)(s * 0.25f);
}

// Exact k-th largest of each score row (4096) via binary search on monotone keys.
__global__ void k_thresh(const float* __restrict__ scores, const int* __restrict__ kptr,
                         float* __restrict__ th) {
  __shared__ uint32_t keys[FD_];
  __shared__ int red[256];
  int row = blockIdx.x, tid = threadIdx.x;
  const float* s = scores + (size_t)row * FD_;
  for (int ii = tid; ii < FD_; ii += 256) {
    uint32_t bb = __float_as_uint(s[ii]);
    keys[ii] = (bb & 0x80000000u) ? ~bb : (bb | 0x80000000u);
  }
  __syncthreads();
  int k = *kptr;
  unsigned long long lo = 0ull, hi = 0xFFFFFFFFull;
  while (lo < hi) {
    unsigned long long mid = (lo + hi + 1ull) >> 1;
    uint32_t m32 = (uint32_t)mid;
    int c = 0;
    for (int ii = tid; ii < FD_; ii += 256) c += (keys[ii] >= m32) ? 1 : 0;
    red[tid] = c; __syncthreads();
    for (int st = 128; st > 0; st >>= 1) { if (tid < st) red[tid] += red[tid + st]; __syncthreads(); }
    c = red[0]; __syncthreads();
    if (c >= k) lo = mid; else hi = mid - 1ull;
  }
  if (tid == 0) {
    uint32_t kk = (uint32_t)lo;
    uint32_t bb = (kk & 0x80000000u) ? (kk & 0x7FFFFFFFu) : ~kk;
    th[row] = __uint_as_float(bb);
  }
}

// ---------------------------------------------------------------------------
// WMMA GEMM 1: fused gate/up/SiLU/mask.
// grid (FD/64, M/64); 4 waves stacked in M; WG tile 64(m) x 64(n).
// Both weight tiles (shared by all 4 waves) are async-staged into LDS,
// double buffered: rows 0..63 = Wg[n0..n0+63], rows 64..127 = Wu[n0..n0+63].
// Output: mout[m][n] = hard ? 0 : -act   (delta = mout @ Wd^T)
// ---------------------------------------------------------------------------
__global__ void __launch_bounds__(128) k_gateup(
    const float* __restrict__ h, const float* __restrict__ wg,
    const float* __restrict__ wu, const float* __restrict__ scores,
    const float* __restrict__ th, float* __restrict__ mout) {
  __shared__ float bsh[2][128][32];
  const int tid = threadIdx.x;
  const int lane = threadIdx.x & 31;
  const int wv = threadIdx.x >> 5;
  const int l15 = lane & 15, kh = lane >> 4;
  const int m0 = blockIdx.y * 64 + wv * 16;
  const int n0 = blockIdx.x * 64;
  const float* ap = h + (size_t)(m0 + l15) * D_ + 2 * kh;

  auto stage = [&](int buf, int kc) {
#pragma unroll
    for (int it = 0; it < 8; ++it) {
      int c = it * 128 + tid;          // 1024 chunks of 16B
      int row = c >> 3, i4 = (c & 7) * 4;
      const float* src = (row < 64 ? wg + (size_t)(n0 + row) * D_
                                   : wu + (size_t)(n0 + row - 64) * D_) + kc + i4;
      async_cp16(&bsh[buf][row][i4], src);
    }
  };

  v8f ag[4] = {};
  v8f au[4] = {};
  stage(0, 0);
  async_wait0();
  __syncthreads();
  int buf = 0;
  for (int kc = 0; kc < D_; kc += 32) {
    if (kc + 32 < D_) stage(buf ^ 1, kc + 32);
    __builtin_prefetch(ap + kc + 256, 0, 1);
#pragma unroll
    for (int kk = 0; kk < 32; kk += 4) {
      v2f a = *(const v2f*)(ap + kc + kk);
#pragma unroll
      for (int t = 0; t < 4; ++t) {
        v2f bg = *(const v2f*)(&bsh[buf][t * 16 + l15][kk + 2 * kh]);
        ag[t] = wmma_f32(a, bg, ag[t]);
        v2f bu = *(const v2f*)(&bsh[buf][64 + t * 16 + l15][kk + 2 * kh]);
        au[t] = wmma_f32(a, bu, au[t]);
      }
    }
    async_wait0();
    __syncthreads();
    buf ^= 1;
  }
  // whole 16-row wave tile lies in one token block -> one threshold per wave
  const int b = m0 >> 10;
  const int tb = (m0 & 1023) >> 4;
  const int srow = b * NBLK + tb;
  const float thv = th[srow];
  const float* sp = scores + (size_t)srow * FD_;
#pragma unroll
  for (int t = 0; t < 4; ++t) {
    const int n = n0 + t * 16 + l15;
    const float keep = (sp[n] >= thv) ? 0.0f : 1.0f;  // selected -> delta contrib 0
#pragma unroll
    for (int r = 0; r < 8; ++r) {
      const int m = m0 + r + 8 * kh;
      float g = ag[t][r], u = au[t][r];
      float act = g * sigmf(g) * u;
      mout[(size_t)m * FD_ + n] = -act * keep;
    }
  }
}

// ---------------------------------------------------------------------------
// WMMA GEMM core, wave tile 32x64, 4 waves as 2(M)x2(N) -> WG tile 64x128.
// A tile (64 rows) and B tile (128 rows) async-staged to LDS, double buffered:
// LDS rows 0..63 = A[mb..mb+63], rows 64..191 = W[nb..nb+127].
// EPI==1: h_out = dh_next + (1-ss)*acc ; EPI==0: plain store (lm_head).
// ---------------------------------------------------------------------------
template <int KA, int EPI>
__global__ void __launch_bounds__(128) k_gemm32x64(
    const float* __restrict__ A, const float* __restrict__ W,
    float* __restrict__ out, const float* __restrict__ addv,
    const float* __restrict__ ssv, int li, int ldo) {
  __shared__ float sh[2][192][32];
  const int tid = threadIdx.x;
  const int lane = threadIdx.x & 31;
  const int wv = threadIdx.x >> 5;
  const int wm = wv >> 1, wn = wv & 1;
  const int l15 = lane & 15, kh = lane >> 4;
  const int mb = blockIdx.y * 64;   // A tile base row
  const int nb = blockIdx.x * 128;  // B tile base row
  const int m0 = mb + wm * 32;
  const int n0loc = wn * 64;

  auto stage = [&](int buf, int kc) {
#pragma unroll
    for (int it = 0; it < 12; ++it) {
      int c = it * 128 + tid;          // 1536 chunks of 16B
      int row = c >> 3, i4 = (c & 7) * 4;
      const float* src = (row < 64 ? A + (size_t)(mb + row) * KA
                                   : W + (size_t)(nb + row - 64) * KA) + kc + i4;
      async_cp16(&sh[buf][row][i4], src);
    }
  };

  v8f acc[2][4] = {};
  stage(0, 0);
  async_wait0();
  __syncthreads();
  int buf = 0;
  for (int kc = 0; kc < KA; kc += 32) {
    if (kc + 32 < KA) stage(buf ^ 1, kc + 32);
#pragma unroll
    for (int kk = 0; kk < 32; kk += 4) {
      v2f av0 = *(const v2f*)(&sh[buf][wm * 32 + l15][kk + 2 * kh]);
      v2f av1 = *(const v2f*)(&sh[buf][wm * 32 + 16 + l15][kk + 2 * kh]);
#pragma unroll
      for (int t = 0; t < 4; ++t) {
        v2f bv = *(const v2f*)(&sh[buf][64 + n0loc + t * 16 + l15][kk + 2 * kh]);
        acc[0][t] = wmma_f32(av0, bv, acc[0][t]);
        acc[1][t] = wmma_f32(av1, bv, acc[1][t]);
      }
    }
    async_wait0();
    __syncthreads();
    buf ^= 1;
  }
#pragma unroll
  for (int mt = 0; mt < 2; ++mt) {
#pragma unroll
    for (int t = 0; t < 4; ++t) {
      const int n = nb + n0loc + t * 16 + l15;
#pragma unroll
      for (int r = 0; r < 8; ++r) {
        const int m = m0 + mt * 16 + r + 8 * kh;
        float v = acc[mt][t][r];
        if (EPI == 1) {
          float s = ssv[(m >> 10) * L_ + li];
          out[(size_t)m * ldo + n] = addv[(size_t)m * ldo + n] + (1.0f - s) * v;
        } else {
          out[(size_t)m * ldo + n] = v;
        }
      }
    }
  }
}

// ---------------------------------------------------------------------------
// Host side: input resolution + launch graph
// ---------------------------------------------------------------------------

enum {
  I_DENSE, I_GATE, I_UP, I_DOWN, I_LM,
  I_CHW, I_CHB, I_BHW, I_BHB, I_IPW, I_IPB, I_LQ, I_P1W, I_P1B, I_P2W, I_P2B,
  I_RBW, I_RBB, I_RCW, I_RCB, I_RHW, I_RHB, I_RLNB, I_RLNG, I_RLW, I_RLB,
  I_RT1W, I_RT1B, I_RT2W, I_RT2B, I_SKW, I_SKB, I_S1W, I_S1B, I_S2W, I_S2B,
  N_SEM
};

static const long long sem_size[N_SEM] = {
  71303168LL, 29360128LL, 29360128LL, 29360128LL, 32768000LL,
  16384, 64, 256, 1, 262144, 256, 4096, 131072, 256, 65536, 256,
  1792, 896, 57344, 896, 3670016, 28672, 7168, 7168, 917504, 896,
  344064, 896, 114688, 896, 256, 1, 196608, 256, 65536, 256
};

static const unsigned char params_sorted_[31] = {
  I_CHW, I_CHB, I_BHW, I_BHB, I_IPW, I_IPB, I_LQ, I_P1W, I_P1B, I_P2W, I_P2B,
  I_RBW, I_RBB, I_RCW, I_RCB, I_RHW, I_RHB, I_RLNB, I_RLNG, I_RLW, I_RLB,
  I_RT1W, I_RT1B, I_RT2W, I_RT2B, I_SKW, I_SKB, I_S1W, I_S1B, I_S2W, I_S2B
};
static const unsigned char params_ins_[31] = {
  I_IPW, I_IPB, I_S1W, I_S1B, I_S2W, I_S2B, I_LQ, I_P1W, I_P1B, I_P2W, I_P2B,
  I_BHW, I_BHB, I_SKW, I_SKB, I_CHW, I_CHB, I_RLNG, I_RLNB, I_RLW, I_RLB,
  I_RCW, I_RCB, I_RBW, I_RBB, I_RT1W, I_RT1B, I_RT2W, I_RT2B, I_RHW, I_RHB
};

static bool try_layout(void* const* d_in, const int* in_sizes, int n_in,
                       const unsigned char* order, const float** tab) {
  if (n_in != N_SEM) return false;
  for (int idx = 0; idx < N_SEM; ++idx)
    if ((long long)in_sizes[idx] != sem_size[order[idx]]) return false;
  for (int idx = 0; idx < N_SEM; ++idx) tab[order[idx]] = (const float*)d_in[idx];
  return true;
}

extern "C" void kernel_launch(void* const* d_in, const int* in_sizes, int n_in,
                              void* d_out, int out_size, void* d_ws, size_t ws_size,
                              hipStream_t stream) {
  (void)out_size; (void)ws_size;
  // ---- resolve input pointers (three plausible pytree flatten orders) ----
  unsigned char LA[N_SEM], LB[N_SEM], LC[N_SEM];
  { int p = 0;
    LA[p++] = I_DENSE; LA[p++] = I_DOWN; LA[p++] = I_GATE; LA[p++] = I_UP; LA[p++] = I_LM;
    for (int q = 0; q < 31; ++q) LA[p++] = params_sorted_[q]; }
  { int p = 0;
    LB[p++] = I_DENSE;
    for (int q = 0; q < 31; ++q) LB[p++] = params_sorted_[q];
    LB[p++] = I_GATE; LB[p++] = I_UP; LB[p++] = I_DOWN; LB[p++] = I_LM; }
  { int p = 0;
    LC[p++] = I_DENSE;
    for (int q = 0; q < 31; ++q) LC[p++] = params_ins_[q];
    LC[p++] = I_GATE; LC[p++] = I_UP; LC[p++] = I_DOWN; LC[p++] = I_LM; }

  const float* P[N_SEM] = {};
  bool ok = try_layout(d_in, in_sizes, n_in, LB, P) ||
            try_layout(d_in, in_sizes, n_in, LA, P) ||
            try_layout(d_in, in_sizes, n_in, LC, P);
  if (!ok) {
    if (n_in != N_SEM) return;
    for (int idx = 0; idx < N_SEM; ++idx) P[LB[idx]] = (const float*)d_in[idx];
  }

  // ---- workspace layout (floats) ----
  float* w = (float*)d_ws;
  size_t off = 0;
  auto alloc = [&](size_t n) { size_t r = off; off += (n + 63) & ~(size_t)63; return r; };
  const size_t WS_SUM  = alloc(3 * B_ * D_);
  const size_t WS_PCAT = alloc(B_ * 3 * CD_);
  const size_t WS_G1   = alloc(B_ * CD_);
  const size_t WS_G    = alloc(B_ * CD_);
  const size_t WS_HCAT = alloc(B_ * L_ * 2 * CD_);
  const size_t WS_P1   = alloc(B_ * L_ * CD_);
  const size_t WS_P2   = alloc(B_ * L_ * CD_);
  const size_t WS_FB   = alloc(B_ * L_);
  const size_t WS_SS   = alloc(B_ * L_);
  const size_t WS_CV   = alloc(B_ * L_ * C_);
  const size_t WS_HP   = alloc((size_t)B_ * NBLK * D_);
  const size_t WS_XC   = alloc(B_ * RD_);
  const size_t WS_XB   = alloc(B_ * RD_);
  const size_t WS_XL   = alloc((size_t)B_ * NBLK * RD_);
  const size_t WS_T1   = alloc((size_t)B_ * NBLK * RD_);
  const size_t WS_T2   = alloc((size_t)B_ * NBLK * RD_);
  const size_t WS_SC   = alloc((size_t)B_ * NBLK * FD_);
  const size_t WS_K    = alloc(64);
  const size_t WS_TH   = alloc(B_ * NBLK);
  const size_t WS_H    = alloc((size_t)M_ * D_);
  const size_t WS_M    = alloc((size_t)M_ * FD_);

  const size_t SLAB = (size_t)M_ * D_;  // one dense_hidden layer

  // ---- controller (tiny) ----
  k_summaries<<<(3 * B_ * D_) / 256, 256, 0, stream>>>(P[I_DENSE], w + WS_SUM);
  k_proj_concat<<<(B_ * 3 * CD_ + 255) / 256, 256, 0, stream>>>(
      w + WS_SUM, P[I_IPW], P[I_IPB], w + WS_PCAT);
  k_linear<<<(B_ * CD_ + 255) / 256, 256, 0, stream>>>(
      w + WS_PCAT, 3 * CD_, P[I_S1W], P[I_S1B], w + WS_G1, B_, 3 * CD_, CD_, 1);
  k_linear<<<(B_ * CD_ + 255) / 256, 256, 0, stream>>>(
      w + WS_G1, CD_, P[I_S2W], P[I_S2B], w + WS_G, B_, CD_, CD_, 0);
  k_hcat<<<(B_ * L_ * 2 * CD_ + 255) / 256, 256, 0, stream>>>(w + WS_G, P[I_LQ], w + WS_HCAT);
  k_linear<<<(B_ * L_ * CD_ + 255) / 256, 256, 0, stream>>>(
      w + WS_HCAT, 2 * CD_, P[I_P1W], P[I_P1B], w + WS_P1, B_ * L_, 2 * CD_, CD_, 1);
  k_linear<<<(B_ * L_ * CD_ + 255) / 256, 256, 0, stream>>>(
      w + WS_P1, CD_, P[I_P2W], P[I_P2B], w + WS_P2, B_ * L_, CD_, CD_, 1);
  k_linear<<<(B_ * L_ + 255) / 256, 256, 0, stream>>>(
      w + WS_P2, CD_, P[I_BHW], P[I_BHB], w + WS_FB, B_ * L_, CD_, 1, 2);
  k_linear<<<(B_ * L_ + 255) / 256, 256, 0, stream>>>(
      w + WS_P2, CD_, P[I_SKW], P[I_SKB], w + WS_SS, B_ * L_, CD_, 1, 2);
  k_linear<<<(B_ * L_ * C_ + 255) / 256, 256, 0, stream>>>(
      w + WS_P2, CD_, P[I_CHW], P[I_CHB], w + WS_CV, B_ * L_, CD_, C_, 0);

  // ---- sparse FFN layers ----
  for (int j = 0; j < NR_; ++j) {
    const int i = DP_ + j;
    const float* hcur = (j == 0) ? (P[I_DENSE] + (size_t)DP_ * SLAB) : (w + WS_H);

    k_pool<<<(B_ * NBLK * D_) / 256, 256, 0, stream>>>(hcur, w + WS_HP);
    k_ln_local<<<B_ * NBLK, 128, 0, stream>>>(
        w + WS_HP, P[I_RLNG] + (size_t)j * D_, P[I_RLNB] + (size_t)j * D_,
        P[I_RLW] + (size_t)j * RD_ * D_, P[I_RLB] + (size_t)j * RD_, w + WS_XL);
    k_router_vec<<<(B_ * RD_ + 255) / 256, 256, 0, stream>>>(
        w + WS_CV, w + WS_FB, w + WS_SS,
        P[I_RCW] + (size_t)j * RD_ * C_, P[I_RCB] + (size_t)j * RD_,
        P[I_RBW] + (size_t)j * RD_ * 2, P[I_RBB] + (size_t)j * RD_,
        i, w + WS_XC, w + WS_XB);
    k_trunk1<<<(B_ * NBLK * RD_ + 255) / 256, 256, 0, stream>>>(
        w + WS_XL, w + WS_XC, w + WS_XB,
        P[I_RT1W] + (size_t)j * RD_ * 3 * RD_, P[I_RT1B] + (size_t)j * RD_, w + WS_T1);
    k_linear<<<(B_ * NBLK * RD_ + 255) / 256, 256, 0, stream>>>(
        w + WS_T1, RD_, P[I_RT2W] + (size_t)j * RD_ * RD_, P[I_RT2B] + (size_t)j * RD_,
        w + WS_T2, B_ * NBLK, RD_, RD_, 1);
    k_linear<<<(B_ * NBLK * FD_ + 255) / 256, 256, 0, stream>>>(
        w + WS_T2, RD_, P[I_RHW] + (size_t)j * FD_ * RD_, P[I_RHB] + (size_t)j * FD_,
        w + WS_SC, B_ * NBLK, RD_, FD_, 0);
    k_kcalc<<<1, 1, 0, stream>>>(w + WS_FB, i, (int*)(w + WS_K));
    k_thresh<<<B_ * NBLK, 256, 0, stream>>>(w + WS_SC, (const int*)(w + WS_K), w + WS_TH);

    k_gateup<<<dim3(FD_ / 64, M_ / 64), 128, 0, stream>>>(
        hcur, P[I_GATE] + (size_t)j * FD_ * D_, P[I_UP] + (size_t)j * FD_ * D_,
        w + WS_SC, w + WS_TH, w + WS_M);
    k_gemm32x64<FD_, 1><<<dim3(D_ / 128, M_ / 64), 128, 0, stream>>>(
        w + WS_M, P[I_DOWN] + (size_t)j * D_ * FD_, w + WS_H,
        P[I_DENSE] + (size_t)(i + 1) * SLAB, w + WS_SS, i, D_);
  }

  // ---- LM head ----
  k_gemm32x64<D_, 0><<<dim3(V_ / 128, M_ / 64), 128, 0, stream>>>(
      w + WS_H, P[I_LM], (float*)d_out, nullptr, nullptr, 0, V_);
}